// GNNModel_54829552501389
// MI455X (gfx1250) — compile-verified
//
#include <hip/hip_runtime.h>
#include <hip/hip_bf16.h>

// ---------------------------------------------------------------------------
// GCN: 2-layer graph conv, 50000 nodes, 800000 edges, feats 64 -> 128 -> 64.
// GEMMs use V_WMMA_F32_16X16X4_F32 (f32 in / f32 acc, exact precision match).
// Edge aggregation via global f32 atomics; gathers vectorized to b128
// (working sets ~13-26MB each, all resident in the 192MB L2).
// ---------------------------------------------------------------------------

#define N_NODES 50000
#define N_EDGES 800000
#define IN_F    64
#define HID     128
#define NCLS    64

typedef __attribute__((ext_vector_type(2))) float v2f;
typedef __attribute__((ext_vector_type(8))) float v8f;

// --------------------------- zeroing (float4) ------------------------------
__global__ __launch_bounds__(256) void gnn_zero3(float4* a, long long na4,
                                                 float4* b, long long nb4,
                                                 float4* c, long long nc4) {
    long long stride = (long long)gridDim.x * blockDim.x;
    const float4 z = make_float4(0.f, 0.f, 0.f, 0.f);
    for (long long i = (long long)blockIdx.x * blockDim.x + threadIdx.x; i < na4; i += stride) a[i] = z;
    for (long long i = (long long)blockIdx.x * blockDim.x + threadIdx.x; i < nb4; i += stride) b[i] = z;
    for (long long i = (long long)blockIdx.x * blockDim.x + threadIdx.x; i < nc4; i += stride) c[i] = z;
}

// --------------------------- degrees --------------------------------------
__global__ __launch_bounds__(256) void gnn_degrees(const int* __restrict__ src,
                                                   const int* __restrict__ dst,
                                                   float* __restrict__ in_deg,   // -> norm_in slot
                                                   float* __restrict__ out_deg)  // -> norm_out slot
{
    int e = blockIdx.x * blockDim.x + threadIdx.x;
    if (e >= N_EDGES) return;
    atomicAdd(&in_deg[dst[e]],  1.0f);
    atomicAdd(&out_deg[src[e]], 1.0f);
}

__global__ __launch_bounds__(256) void gnn_norms(float* __restrict__ ni,
                                                 float* __restrict__ no) {
    int i = blockIdx.x * blockDim.x + threadIdx.x;
    if (i >= N_NODES) return;
    ni[i] = 1.0f / sqrtf(fmaxf(ni[i], 1.0f));
    no[i] = 1.0f / sqrtf(fmaxf(no[i], 1.0f));
}

// --------------------------- edge scatter, layer 1 ------------------------
// agg1[dst] += features[src] * norm_in[src] * norm_out[src]
// 16 threads per edge, float4 (b128) gather + 4 atomics each.
__global__ __launch_bounds__(256) void gnn_scatter1(const float4* __restrict__ feat4,
                                                    const int* __restrict__ src,
                                                    const int* __restrict__ dst,
                                                    const float* __restrict__ ni,
                                                    const float* __restrict__ no,
                                                    float* __restrict__ agg1) {
    unsigned idx = blockIdx.x * blockDim.x + threadIdx.x;   // edge*16 + quad
    unsigned e = idx >> 4;
    unsigned q = idx & 15u;
    if (e >= N_EDGES) return;
    int s = src[e], d = dst[e];
    float scale = ni[s] * no[s];
    float4 v = feat4[(long long)s * (IN_F / 4) + q];
    float* dp = agg1 + (long long)d * IN_F + q * 4;
    atomicAdd(dp + 0, v.x * scale);
    atomicAdd(dp + 1, v.y * scale);
    atomicAdd(dp + 2, v.z * scale);
    atomicAdd(dp + 3, v.w * scale);
}

// --------------------------- GEMM1 (WMMA f32 16x16x4) ---------------------
// h1s = relu((agg1 * norm_in) @ W1 + b1) * norm_in * norm_out
// M=50000 (3125 tiles), K=64, N=128 (8 tiles). One 16x16 tile per wave.
__global__ __launch_bounds__(256) void gnn_gemm1_wmma(const float* __restrict__ agg1,
                                                      const float* __restrict__ W1,  // [64,128]
                                                      const float* __restrict__ b1,  // [128]
                                                      const float* __restrict__ ni,
                                                      const float* __restrict__ no,
                                                      float* __restrict__ h1s) {     // [N,128]
    const int lane = threadIdx.x & 31;
    const int wave = threadIdx.x >> 5;
    const int tile = blockIdx.x * (blockDim.x >> 5) + wave;
    const int NT = HID / 16;                  // 8
    const int mTile = tile / NT;
    const int nTile = tile % NT;
    if (mTile >= N_NODES / 16) return;        // wave-uniform

    const int half = lane >> 4;
    const int lp   = lane & 15;
    const int rowA = mTile * 16 + lp;         // A-fragment row for this lane
    const int col  = nTile * 16 + lp;         // B/C/D column for this lane
    const float sA = ni[rowA];

    v8f c = {};
    const float* arow = agg1 + (long long)rowA * IN_F;
#pragma unroll
    for (int k0 = 0; k0 < IN_F; k0 += 4) {
        int ka = k0 + 2 * half;
        v2f a, b;
        a.x = arow[ka]     * sA;
        a.y = arow[ka + 1] * sA;
        b.x = W1[(ka)     * HID + col];
        b.y = W1[(ka + 1) * HID + col];
        c = __builtin_amdgcn_wmma_f32_16x16x4_f32(false, a, false, b,
                                                  (short)0, c, false, false);
    }

    const float bias = b1[col];
#pragma unroll
    for (int v = 0; v < 8; ++v) {
        int row = mTile * 16 + v + 8 * half;  // D-fragment row of element v
        float z = c[v] + bias;
        z = z > 0.0f ? z : 0.0f;
        h1s[(long long)row * HID + col] = z * ni[row] * no[row];
    }
}

// --------------------------- GEMM2 (WMMA f32 16x16x4) ---------------------
// t = h1s @ W2.   M=50000, K=128, N=64 (4 tiles).
__global__ __launch_bounds__(256) void gnn_gemm2_wmma(const float* __restrict__ h1s, // [N,128]
                                                      const float* __restrict__ W2,  // [128,64]
                                                      float* __restrict__ t) {       // [N,64]
    const int lane = threadIdx.x & 31;
    const int wave = threadIdx.x >> 5;
    const int tile = blockIdx.x * (blockDim.x >> 5) + wave;
    const int NT = NCLS / 16;                 // 4
    const int mTile = tile / NT;
    const int nTile = tile % NT;
    if (mTile >= N_NODES / 16) return;        // wave-uniform

    const int half = lane >> 4;
    const int lp   = lane & 15;
    const int rowA = mTile * 16 + lp;
    const int col  = nTile * 16 + lp;

    v8f c = {};
    const float* arow = h1s + (long long)rowA * HID;
#pragma unroll
    for (int k0 = 0; k0 < HID; k0 += 4) {
        int ka = k0 + 2 * half;
        v2f a, b;
        a.x = arow[ka];
        a.y = arow[ka + 1];
        b.x = W2[(ka)     * NCLS + col];
        b.y = W2[(ka + 1) * NCLS + col];
        c = __builtin_amdgcn_wmma_f32_16x16x4_f32(false, a, false, b,
                                                  (short)0, c, false, false);
    }
#pragma unroll
    for (int v = 0; v < 8; ++v) {
        int row = mTile * 16 + v + 8 * half;
        t[(long long)row * NCLS + col] = c[v];
    }
}

// --------------------------- edge scatter, layer 2 ------------------------
// out[dst] += t[src]; 16 threads/edge, b128 gather + 4 atomics.
__global__ __launch_bounds__(256) void gnn_scatter2(const float4* __restrict__ t4,
                                                    const int* __restrict__ src,
                                                    const int* __restrict__ dst,
                                                    float* __restrict__ out) {
    unsigned idx = blockIdx.x * blockDim.x + threadIdx.x;
    unsigned e = idx >> 4;
    unsigned q = idx & 15u;
    if (e >= N_EDGES) return;
    int s = src[e], d = dst[e];
    float4 v = t4[(long long)s * (NCLS / 4) + q];
    float* dp = out + (long long)d * NCLS + q * 4;
    atomicAdd(dp + 0, v.x);
    atomicAdd(dp + 1, v.y);
    atomicAdd(dp + 2, v.z);
    atomicAdd(dp + 3, v.w);
}

// --------------------------- finalize (float4) -----------------------------
__global__ __launch_bounds__(256) void gnn_finalize(float4* __restrict__ out4,
                                                    const float* __restrict__ ni,
                                                    const float4* __restrict__ b24) {
    unsigned idx = blockIdx.x * blockDim.x + threadIdx.x;   // node*16 + quad
    unsigned i = idx >> 4;
    unsigned q = idx & 15u;
    if (i >= N_NODES) return;
    float s = ni[i];
    float4 v = out4[idx];
    float4 b = b24[q];
    v.x = v.x * s + b.x;
    v.y = v.y * s + b.y;
    v.z = v.z * s + b.z;
    v.w = v.w * s + b.w;
    out4[idx] = v;
}

// ---------------------------------------------------------------------------
extern "C" void kernel_launch(void* const* d_in, const int* in_sizes, int n_in,
                              void* d_out, int out_size, void* d_ws, size_t ws_size,
                              hipStream_t stream) {
    const float* features = (const float*)d_in[0];
    const float* W1       = (const float*)d_in[1];
    const float* b1       = (const float*)d_in[2];
    const float* W2       = (const float*)d_in[3];
    const float* b2       = (const float*)d_in[4];
    const int*   src      = (const int*)d_in[5];
    const int*   dst      = (const int*)d_in[6];
    float*       out      = (float*)d_out;

    // workspace layout (floats): norms | agg1/t (N*64) | h1s (N*128)
    float* ws       = (float*)d_ws;
    float* norm_in  = ws;
    float* norm_out = ws + N_NODES;
    float* bufA     = ws + 2 * N_NODES;                             // agg1, later t
    float* bufB     = ws + 2 * N_NODES + (long long)N_NODES * IN_F; // h1s

    // 1) zero norms+agg1+out
    gnn_zero3<<<2048, 256, 0, stream>>>((float4*)norm_in, (2LL * N_NODES) / 4,
                                        (float4*)bufA, ((long long)N_NODES * IN_F) / 4,
                                        (float4*)out, ((long long)N_NODES * NCLS) / 4);
    // 2) degrees
    gnn_degrees<<<(N_EDGES + 255) / 256, 256, 0, stream>>>(src, dst, norm_in, norm_out);
    // 3) norms
    gnn_norms<<<(N_NODES + 255) / 256, 256, 0, stream>>>(norm_in, norm_out);
    // 4) layer-1 aggregation: 16 threads/edge
    {
        long long threads = (long long)N_EDGES * 16;
        gnn_scatter1<<<(unsigned)((threads + 255) / 256), 256, 0, stream>>>(
            (const float4*)features, src, dst, norm_in, norm_out, bufA);
    }
    // 5) GEMM1: (N/16)*(HID/16) = 3125*8 = 25000 tiles, 8 waves/block
    gnn_gemm1_wmma<<<(25000 + 7) / 8, 256, 0, stream>>>(bufA, W1, b1, norm_in, norm_out, bufB);
    // 6) GEMM2: 3125*4 = 12500 tiles (t reuses bufA)
    gnn_gemm2_wmma<<<(12500 + 7) / 8, 256, 0, stream>>>(bufB, W2, bufA);
    // 7) layer-2 aggregation into out: 16 threads/edge
    {
        long long threads = (long long)N_EDGES * 16;
        gnn_scatter2<<<(unsigned)((threads + 255) / 256), 256, 0, stream>>>(
            (const float4*)bufA, src, dst, out);
    }
    // 8) finalize: float4 per thread
    {
        long long threads = (long long)N_NODES * 16;
        gnn_finalize<<<(unsigned)((threads + 255) / 256), 256, 0, stream>>>(
            (float4*)out, norm_in, (const float4*)b2);
    }
}